// PillarLayer_55628416418346
// MI455X (gfx1250) — compile-verified
//
#include <hip/hip_runtime.h>

typedef __attribute__((ext_vector_type(2))) float v2f;
typedef __attribute__((ext_vector_type(4))) float v4f;
typedef __attribute__((ext_vector_type(8))) float v8f;

#define NPTS_ 32
#define CCH_  4
#define PSTRIDE (NPTS_ * CCH_)   // 128 floats per pillar

// One wave32 handles 16 pillars:
//  - coalesced b128 pass-through copy of those pillars (warms cache)
//  - 3 channels x 8 chained V_WMMA_F32_16X16X4_F32 against a ones B-matrix
//    => exact fp32 sum over the 32 points per pillar
//  - lanes 0 / 16 divide by npoints and scatter into the (bs,3,y_l,x_l) canvas
__global__ __launch_bounds__(256) void pillar_center_wmma(
    const float* __restrict__ pillars,
    const int*   __restrict__ coors,
    const int*   __restrict__ npoints,
    const int*   __restrict__ xl_p,
    const int*   __restrict__ yl_p,
    float* __restrict__ out_pillars,
    float* __restrict__ canvas,
    int P)
{
    const int tid  = blockIdx.x * blockDim.x + threadIdx.x;
    const int wave = tid >> 5;
    const int lane = tid & 31;
    const int pb   = wave * 16;
    if (pb >= P) return;   // uniform per wave

    // hint the per-pillar metadata into cache (global_prefetch_b8)
    __builtin_prefetch(coors + 4 * (size_t)pb, 0, 0);
    __builtin_prefetch(npoints + pb, 0, 0);

    // ---- fused pass-through copy: 16 pillars * 128 floats = 512 float4 ----
    {
        const v4f* __restrict__ src = (const v4f*)(pillars + (size_t)pb * PSTRIDE);
        v4f* __restrict__ dst = (v4f*)(out_pillars + (size_t)pb * PSTRIDE);
        const int npil = (P - pb >= 16) ? 16 : (P - pb);
        const int nf4  = (npil * PSTRIDE) >> 2;
        #pragma unroll
        for (int i = 0; i < 16; ++i) {
            const int idx = i * 32 + lane;
            if (idx < nf4) dst[idx] = src[idx];
        }
    }

    // ---- WMMA reduction over the 32 points, per channel (x,y,z) ----
    // A layout (32-bit 16x4, ISA 7.12.2): lane = half*16 + m, m = pillar row,
    // VGPR0 holds K = 2*half, VGPR1 holds K = 2*half+1.
    const int m    = lane & 15;
    const int half = lane >> 4;
    int p = pb + m;
    if (p >= P) p = P - 1;   // clamp (keeps EXEC all-ones for WMMA)
    const float* __restrict__ abase =
        pillars + (size_t)p * PSTRIDE + (size_t)(half * 2) * CCH_;

    v2f bones = {1.0f, 1.0f};     // B = ones(4x16): layout-independent
    v8f acc0 = {}, acc1 = {}, acc2 = {};
    #pragma unroll
    for (int g = 0; g < 8; ++g) {
        const float* gb = abase + g * 4 * CCH_;   // point 4g + 2*half, channel 0
        v2f a0; a0.x = gb[0 * CCH_ + 0]; a0.y = gb[1 * CCH_ + 0];
        v2f a1; a1.x = gb[0 * CCH_ + 1]; a1.y = gb[1 * CCH_ + 1];
        v2f a2; a2.x = gb[0 * CCH_ + 2]; a2.y = gb[1 * CCH_ + 2];
        acc0 = __builtin_amdgcn_wmma_f32_16x16x4_f32(false, a0, false, bones,
                                                     (short)0, acc0, false, false);
        acc1 = __builtin_amdgcn_wmma_f32_16x16x4_f32(false, a1, false, bones,
                                                     (short)0, acc1, false, false);
        acc2 = __builtin_amdgcn_wmma_f32_16x16x4_f32(false, a2, false, bones,
                                                     (short)0, acc2, false, false);
    }

    // ---- epilogue: D VGPR r on lane-half h holds pillar pb + 8*h + r ----
    if ((lane & 15) == 0) {
        const int xl = *xl_p;
        const int yl = *yl_p;
        const size_t plane = (size_t)xl * (size_t)yl;
        const int pbase = pb + half * 8;
        #pragma unroll
        for (int r = 0; r < 8; ++r) {
            const int pp = pbase + r;
            if (pp < P) {
                const float cnt = (float)npoints[pp];
                const int bb = coors[4 * pp + 0];
                const int xx = coors[4 * pp + 1];
                const int yy = coors[4 * pp + 2];
                // out[b, c, y, x] = center[p][c]
                const size_t ob = ((size_t)bb * 3) * plane + (size_t)yy * xl + xx;
                canvas[ob]             = acc0[r] / cnt;
                canvas[ob + plane]     = acc1[r] / cnt;
                canvas[ob + 2 * plane] = acc2[r] / cnt;
            }
        }
    }
}

__global__ void fzero_kernel(float* __restrict__ p, size_t n) {
    size_t i = (size_t)blockIdx.x * blockDim.x + threadIdx.x;
    if (i < n) p[i] = 0.0f;
}

__global__ void i2f_kernel(const int* __restrict__ s, float* __restrict__ d, int n) {
    int i = blockIdx.x * blockDim.x + threadIdx.x;
    if (i < n) d[i] = (float)s[i];
}

extern "C" void kernel_launch(void* const* d_in, const int* in_sizes, int n_in,
                              void* d_out, int out_size, void* d_ws, size_t ws_size,
                              hipStream_t stream) {
    const float* pillars = (const float*)d_in[0];
    const int*   coors   = (const int*)d_in[1];
    const int*   npoints = (const int*)d_in[2];
    // d_in[3] = bs (unused on device; canvas extent derived from out_size)
    const int*   xl_p    = (const int*)d_in[4];
    const int*   yl_p    = (const int*)d_in[5];
    float* out = (float*)d_out;

    const size_t n0   = (size_t)in_sizes[0];           // pillars floats
    const size_t n1   = (size_t)in_sizes[1];           // coors ints
    const size_t n2   = (size_t)in_sizes[2];           // npoints ints
    const size_t off3 = n0 + n1 + n2;                  // canvas offset in d_out
    const size_t n3   = (size_t)out_size - off3;       // canvas floats (bs*3*y_l*x_l)
    const int    P    = in_sizes[2];                   // number of pillars

    // 1) zero the canvas region (d_out is poisoned / stale between replays)
    {
        unsigned blocks = (unsigned)((n3 + 255) / 256);
        fzero_kernel<<<blocks, 256, 0, stream>>>(out + off3, n3);
    }
    // 2) int -> float pass-through outputs
    i2f_kernel<<<(in_sizes[1] + 255) / 256, 256, 0, stream>>>(coors,   out + n0,      in_sizes[1]);
    i2f_kernel<<<(in_sizes[2] + 255) / 256, 256, 0, stream>>>(npoints, out + n0 + n1, in_sizes[2]);
    // 3) fused copy + WMMA reduction + scatter (after the zero-fill on this stream)
    {
        int waves   = (P + 15) / 16;
        long long threads = (long long)waves * 32;
        unsigned blocks = (unsigned)((threads + 255) / 256);
        pillar_center_wmma<<<blocks, 256, 0, stream>>>(
            pillars, coors, npoints, xl_p, yl_p, out, out + off3, P);
    }
}